// Fature_embedding_59072980189461
// MI455X (gfx1250) — compile-verified
//
#include <hip/hip_runtime.h>
#include <stdint.h>

#define BATCH   4096
#define FIELDS  16
#define VOCAB   500000
#define DIM     32
#define NPAIR   120                                  // 16 choose 2
#define PAIR_FLOATS (NPAIR * (DIM + 1))              // 3960
#define DIAG_FLOATS (FIELDS * DIM)                   // 512
#define OUT_PER_B   (PAIR_FLOATS + DIAG_FLOATS + FIELDS)  // 4488
#define THREADS 256

#if defined(__HIP_DEVICE_COMPILE__)
#if __has_builtin(__builtin_amdgcn_global_load_async_to_lds_b128) && \
    __has_builtin(__builtin_amdgcn_s_wait_asynccnt)
#define USE_ASYNC_LDS 1
#endif
#endif

typedef int v4i __attribute__((vector_size(4 * sizeof(int))));
typedef __attribute__((address_space(1))) v4i as1_v4i;
typedef __attribute__((address_space(3))) v4i as3_v4i;

__global__ __launch_bounds__(THREADS) void ffm_embed_kernel(
    const int* __restrict__ x, const float* __restrict__ W,
    const float* __restrict__ Lw, float* __restrict__ out)
{
    // g[(i*16 + j)*32 + d] = W[j, x_b[i], d]   (32 KB)
    __shared__ float g[FIELDS * FIELDS * DIM];
    __shared__ float innerv[NPAIR];             // pairwise inner products
    __shared__ unsigned short pair_ij[NPAIR];   // (i<<8)|j, i<j, triu row-major
    __shared__ int xs[FIELDS];

    const int b   = blockIdx.x;
    const int tid = threadIdx.x;
    const int* xb = x + (size_t)b * FIELDS;

    if (tid < FIELDS) xs[tid] = xb[tid];
    if (tid < NPAIR) {
        int i = 0, rem = tid;
        while (rem >= FIELDS - 1 - i) { rem -= FIELDS - 1 - i; ++i; }
        int j = i + 1 + rem;
        pair_ij[tid] = (unsigned short)((i << 8) | j);
    }
    __syncthreads();   // xs + pair table visible

    // ---- Stage 1: gather 256 embedding rows (128 B each) into LDS ----
    // 2048 lane-tasks, each moves one 16-byte chunk: row = task>>3, chunk = task&7
#pragma unroll
    for (int k = tid; k < FIELDS * FIELDS * (DIM / 4); k += THREADS) {
        const int row = k >> 3;
        const int c   = k & 7;
        const int i   = row >> 4;
        const int j   = row & 15;
        const size_t gidx = ((size_t)j * VOCAB + (size_t)xs[i]) * DIM + (size_t)(c * 4);
        float* ldst = &g[row * DIM + c * 4];
#ifdef USE_ASYNC_LDS
        as1_v4i* src = (as1_v4i*)(uintptr_t)(const void*)(W + gidx);
        as3_v4i* dst = (as3_v4i*)(uint32_t)(uintptr_t)(void*)ldst;
        __builtin_amdgcn_global_load_async_to_lds_b128(src, dst, 0, 0);
#else
        *(float4*)ldst = *(const float4*)(W + gidx);
#endif
    }
#ifdef USE_ASYNC_LDS
    __builtin_amdgcn_s_wait_asynccnt(0);
#endif
    __syncthreads();

    // ---- Stage 2a: cooperative inner products (8 lanes per pair) ----
    // 960 tasks in exact 8-lane groups; group-local reduce via lane shuffles
    for (int k = tid; k < NPAIR * 8; k += THREADS) {
        const int p  = k >> 3;
        const int c  = k & 7;
        const int ij = pair_ij[p];
        const int i  = (ij >> 8) & 15;
        const int j  = ij & 15;
        const float4 av = *(const float4*)&g[(i * FIELDS + j) * DIM + c * 4];
        const float4 bv = *(const float4*)&g[(j * FIELDS + i) * DIM + c * 4];
        float s = av.x * bv.x + av.y * bv.y + av.z * bv.z + av.w * bv.w;
        s += __shfl_xor(s, 1);
        s += __shfl_xor(s, 2);
        s += __shfl_xor(s, 4);
        if (c == 0) innerv[p] = s;
    }
    __syncthreads();

    // ---- Stage 2b: coalesced production of all 4488 output scalars ----
    const size_t base = (size_t)b * OUT_PER_B;
    for (int o = tid; o < OUT_PER_B; o += THREADS) {
        float v;
        if (o < PAIR_FLOATS) {
            const int p  = o / (DIM + 1);
            const int r  = o - p * (DIM + 1);
            const int ij = pair_ij[p];
            const int i  = (ij >> 8) & 15;
            const int j  = ij & 15;
            if (r < DIM) {
                v = g[(i * FIELDS + j) * DIM + r] * g[(j * FIELDS + i) * DIM + r];
            } else {
                v = innerv[p];
            }
        } else if (o < PAIR_FLOATS + DIAG_FLOATS) {
            const int q = o - PAIR_FLOATS;
            const int f = q >> 5;
            const int d = q & 31;
            v = g[(f * FIELDS + f) * DIM + d];             // W[f, x_f, d]
        } else {
            const int f = o - (PAIR_FLOATS + DIAG_FLOATS);
            v = Lw[xs[f]];
        }
        out[base + o] = v;
    }
}

extern "C" void kernel_launch(void* const* d_in, const int* in_sizes, int n_in,
                              void* d_out, int out_size, void* d_ws, size_t ws_size,
                              hipStream_t stream) {
    const int*   x   = (const int*)d_in[0];
    const float* W   = (const float*)d_in[1];
    const float* Lw  = (const float*)d_in[2];
    float*       out = (float*)d_out;
    (void)in_sizes; (void)n_in; (void)out_size; (void)d_ws; (void)ws_size;
    hipLaunchKernelGGL(ffm_embed_kernel, dim3(BATCH), dim3(THREADS), 0, stream,
                       x, W, Lw, out);
}